// DetectionLoss_39341900431997
// MI455X (gfx1250) — compile-verified
//
#include <hip/hip_runtime.h>
#include <stdint.h>
#include <math.h>

// Problem constants (match reference)
#define Bc   8
#define Ac   9
#define Cnum 21
#define HWc  10000
#define Nc   (Ac * HWc)          // 90000 anchors per image
#define Mc   32                  // target boxes per image
#define TPB  256                 // 8 wave32 waves per block
#define NBLK ((Nc + TPB - 1) / TPB)  // 352 blocks per image

typedef float v2f __attribute__((ext_vector_type(2)));
typedef float v8f __attribute__((ext_vector_type(8)));
typedef int   a4i __attribute__((vector_size(16)));   // 128-bit payload type for async builtin

// Exact fp32 32-lane wave reduction via V_WMMA_F32_16X16X4_F32.
// A = (x, 0) per lane -> A[m,0]=x[m], A[m,2]=x[m+16]; B = ones.
// D[m,n] = x[m] + x[m+16] (same for every n). Lane<16 holds D rows 0..7,
// lane>=16 holds rows 8..15; sum the 8 dwords then fold halves with one
// shfl_xor(16). Requires EXEC all-ones: called unconditionally from full waves.
__device__ __forceinline__ float wave_sum32(float x) {
  v2f a; a[0] = x;    a[1] = 0.0f;
  v2f o; o[0] = 1.0f; o[1] = 1.0f;
  v8f c = {};
  v8f d = __builtin_amdgcn_wmma_f32_16x16x4_f32(
      /*neg_a=*/false, a, /*neg_b=*/false, o,
      /*c_mod=*/(short)0, c, /*reuse_a=*/false, /*reuse_b=*/false);
  float s = ((d[0] + d[1]) + (d[2] + d[3])) + ((d[4] + d[5]) + (d[6] + d[7]));
  s += __shfl_xor(s, 16, 32);
  return s;
}

__device__ __forceinline__ float smooth_l1(float d) {
  float ad = fabsf(d);
  return (ad < 1.0f) ? 0.5f * d * d : ad - 0.5f;
}

__global__ __launch_bounds__(TPB) void det_loss_main(
    const float* __restrict__ cls, const float* __restrict__ reg,
    const float* __restrict__ anchors, const float* __restrict__ tboxes,
    const int* __restrict__ tlabels, float* __restrict__ ws) {
  __shared__ float4 s_tb[Mc];
  __shared__ int    s_lab[Mc];
  __shared__ float  s_wave[TPB / 32][5];

  const int tid = threadIdx.x;
  const int b   = blockIdx.y;
  const int n   = blockIdx.x * TPB + tid;

  // Stage the 512B of target boxes for image b into LDS via the async path.
#if __has_builtin(__builtin_amdgcn_global_load_async_to_lds_b128)
  if (tid < Mc) {
    __attribute__((address_space(1))) a4i* g =
        (__attribute__((address_space(1))) a4i*)(tboxes + (size_t)b * Mc * 4 + (size_t)tid * 4);
    __attribute__((address_space(3))) a4i* l =
        (__attribute__((address_space(3))) a4i*)&s_tb[tid];
    __builtin_amdgcn_global_load_async_to_lds_b128(g, l, 0, 0);
  }
#if __has_builtin(__builtin_amdgcn_s_wait_asynccnt)
  __builtin_amdgcn_s_wait_asynccnt(0);
#else
  asm volatile("s_wait_asynccnt 0" ::: "memory");
#endif
#else
  if (tid < Mc * 4) {
    ((float*)s_tb)[tid] = tboxes[(size_t)b * Mc * 4 + tid];
  }
#endif
  if (tid >= 64 && tid < 64 + Mc) {
    s_lab[tid - 64] = tlabels[b * Mc + (tid - 64)];
  }
  __syncthreads();

  float acc0 = 0.0f, acc1 = 0.0f, acc2 = 0.0f, acc3 = 0.0f, acc4 = 0.0f;

  if (n < Nc) {
    const int a  = n / HWc;
    const int hw = n - a * HWc;

    // Anchor (contiguous float4, coalesced)
    const float4 anc = ((const float4*)anchors)[n];
    const float aw  = anc.z - anc.x;
    const float ah  = anc.w - anc.y;
    const float acx = anc.x + 0.5f * aw;
    const float acy = anc.y + 0.5f * ah;

    // reg components: [(b*A+a), k, hw] -> stride HWc in k, coalesced in hw
    const size_t rb = (size_t)(b * Ac + a) * 4 * HWc + hw;
    const float r0 = reg[rb];
    const float r1 = reg[rb + 1 * HWc];
    const float r2 = reg[rb + 2 * HWc];
    const float r3 = reg[rb + 3 * HWc];

    // Decode box
    const float cx = acx + (r0 * 2.0f - 1.0f) * aw * 0.25f;
    const float cy = acy + (r1 * 2.0f - 1.0f) * ah * 0.25f;
    const float w  = aw * expf(r2);
    const float h  = ah * expf(r3);
    const float dx0 = cx - 0.5f * w, dy0 = cy - 0.5f * h;
    const float dx1 = cx + 0.5f * w, dy1 = cy + 0.5f * h;
    const float areaA = (dx1 - dx0) * (dy1 - dy0);

    // IoU vs 32 target boxes (LDS broadcast), first-max wins ties (JAX argmax)
    float best = -1.0f;
    int   bi   = 0;
    for (int m = 0; m < Mc; ++m) {
      const float4 t = s_tb[m];
      float iw = fminf(dx1, t.z) - fmaxf(dx0, t.x);
      float ih = fminf(dy1, t.w) - fmaxf(dy0, t.y);
      iw = fmaxf(iw, 0.0f);
      ih = fmaxf(ih, 0.0f);
      const float inter = iw * ih;
      const float areaB = (t.z - t.x) * (t.w - t.y);
      const float uni   = areaA + areaB - inter;
      const float iou   = inter / fmaxf(uni, 1e-8f);
      if (iou > best) { best = iou; bi = m; }
    }
    const float posf = (best >= 0.25f) ? 1.0f : 0.0f;
    const float negf = (best <  0.10f) ? 1.0f : 0.0f;
    const int   lab  = s_lab[bi];

    // Stable logsumexp over 21 classes; fully unrolled register array.
    const size_t cb = (size_t)(b * Ac + a) * Cnum * HWc + hw;
    float vals[Cnum];
    float mx = -3.4e38f;
#pragma unroll
    for (int c = 0; c < Cnum; ++c) {
      const float x = cls[cb + (size_t)c * HWc];
      vals[c] = x;
      mx = fmaxf(mx, x);
    }
    float se = 0.0f;
#pragma unroll
    for (int c = 0; c < Cnum; ++c) se += expf(vals[c] - mx);
    const float lse = mx + logf(se);
    float vlab = vals[0];
#pragma unroll
    for (int c = 1; c < Cnum; ++c) vlab = (c == lab) ? vals[c] : vlab;
    const float ce_pos = lse - vlab;
    const float ce_neg = lse - vals[0];

    // Regression target encode + smooth L1
    const float4 gb = s_tb[bi];
    const float gw  = gb.z - gb.x, gh = gb.w - gb.y;
    const float gcx = gb.x + 0.5f * gw, gcy = gb.y + 0.5f * gh;
    const float t0 = ((gcx - acx) * (4.0f / aw) + 1.0f) * 0.5f;
    const float t1 = ((gcy - acy) * (4.0f / ah) + 1.0f) * 0.5f;
    const float t2 = logf(gw / aw);
    const float t3 = logf(gh / ah);
    const float sl = smooth_l1(r0 - t0) + smooth_l1(r1 - t1) +
                     smooth_l1(r2 - t2) + smooth_l1(r3 - t3);

    acc0 = posf;
    acc1 = negf;
    acc2 = ce_pos * posf;
    acc3 = ce_neg * negf;
    acc4 = sl * posf;
  }

  // Exact fp32 per-wave reductions via WMMA (all lanes active here).
  const float w0 = wave_sum32(acc0);
  const float w1 = wave_sum32(acc1);
  const float w2 = wave_sum32(acc2);
  const float w3 = wave_sum32(acc3);
  const float w4 = wave_sum32(acc4);

  const int lane = tid & 31;
  const int wid  = tid >> 5;
  if (lane == 0) {
    s_wave[wid][0] = w0;
    s_wave[wid][1] = w1;
    s_wave[wid][2] = w2;
    s_wave[wid][3] = w3;
    s_wave[wid][4] = w4;
  }
  __syncthreads();

  // Deterministic cross-wave fold; one partial record per block.
  if (tid < 5) {
    float t = 0.0f;
    for (int wv = 0; wv < TPB / 32; ++wv) t += s_wave[wv][tid];
    ws[((size_t)b * NBLK + blockIdx.x) * 5 + tid] = t;
  }
}

__global__ __launch_bounds__(64) void det_loss_finalize(
    const float* __restrict__ ws, float* __restrict__ out) {
  __shared__ float img[Bc][5];
  const int t = threadIdx.x;
  if (t < Bc * 5) {
    const int b = t / 5, i = t % 5;
    float s = 0.0f;
    for (int j = 0; j < NBLK; ++j) s += ws[((size_t)b * NBLK + j) * 5 + i];
    img[b][i] = s;
  }
  __syncthreads();
  if (t == 0) {
    float cls_sum = 0.0f, reg_sum = 0.0f, tpos = 0.0f;
    for (int b = 0; b < Bc; ++b) {
      const float npos = img[b][0], nneg = img[b][1];
      const float spos = img[b][2], sneg = img[b][3], ssl1 = img[b][4];
      const float cls_l = spos / fmaxf(npos, 1.0f) + sneg / fmaxf(nneg, 1.0f);
      const float reg_l = ssl1 / fmaxf(4.0f * npos, 1.0f);
      cls_sum += cls_l;
      reg_sum += reg_l;
      tpos    += npos;
    }
    const float cls_final = cls_sum / (float)Bc;
    const float reg_final = reg_sum / fmaxf(tpos, 1.0f);
    out[0] = cls_final + reg_final;
    out[1] = cls_final;
    out[2] = reg_final;
    out[3] = tpos;
  }
}

extern "C" void kernel_launch(void* const* d_in, const int* in_sizes, int n_in,
                              void* d_out, int out_size, void* d_ws, size_t ws_size,
                              hipStream_t stream) {
  (void)in_sizes; (void)n_in; (void)out_size; (void)ws_size;
  const float* cls     = (const float*)d_in[0];  // [B,A,C,H,W]
  const float* reg     = (const float*)d_in[1];  // [B,A,4,H,W]
  const float* anchors = (const float*)d_in[2];  // [N,4]
  const float* tboxes  = (const float*)d_in[3];  // [B,M,4]
  const int*   tlabels = (const int*)d_in[4];    // [B,M]
  float* ws  = (float*)d_ws;                     // [B*NBLK, 5] partials
  float* out = (float*)d_out;                    // 4 scalars

  dim3 grid(NBLK, Bc);
  det_loss_main<<<grid, TPB, 0, stream>>>(cls, reg, anchors, tboxes, tlabels, ws);
  det_loss_finalize<<<1, 64, 0, stream>>>(ws, out);
}